// HGTLayer_82343112999442
// MI455X (gfx1250) — compile-verified
//
#include <hip/hip_runtime.h>
#include <math.h>

// ---------------- problem constants ----------------
#define N_NODES 24000
#define DIM     128          // IN_DIM == OUT_DIM
#define H_HEADS 8
#define DKK     16
#define T_TYPES 3
#define R_REL   34
#define EPR     10000
#define NT_SZ   (N_NODES / T_TYPES)   // 8000, multiple of 16
#define ETILES  (EPR / 16)            // 625
#define NTILES  (N_NODES / 16)        // 1500

typedef __attribute__((ext_vector_type(2))) float v2f;
typedef __attribute__((ext_vector_type(8))) float v8f;

// FP32 WMMA: D(16x16,f32) = A(16x4,f32) * B(4x16,f32) + C
// A layout (ISA 7.12.2): lane l<16 -> M=l, VGPR0=K0,VGPR1=K1 ; lanes 16-31 -> M=l-16, K2/K3
// C/D layout: VGPR j: lanes 0-15 -> (M=j, N=lane), lanes 16-31 -> (M=j+8, N=lane-16)
__device__ __forceinline__ v8f wmma_f32(v2f a, v2f b, v8f c) {
  return __builtin_amdgcn_wmma_f32_16x16x4_f32(false, a, false, b, (short)0, c,
                                               false, false);
}

__device__ __forceinline__ float sigmoidf_(float x) {
  return 1.0f / (1.0f + expf(-x));
}

// relation -> node-type tables (same formulas as the reference)
__device__ __forceinline__ int src_nt(int r) { return r <= 9 ? 0 : (r <= 21 ? 1 : 2); }
__device__ __forceinline__ int dst_nt(int r) {
  if (r <= 2 || (r >= 10 && r <= 13) || (r >= 22 && r <= 24)) return 0;
  if ((r >= 3 && r <= 6) || (r >= 14 && r <= 17) || (r >= 25 && r <= 28)) return 1;
  return 2;
}

// float atomic max via signed/unsigned int ordering trick (maps to native
// GLOBAL_ATOMIC_MAX_I32 / GLOBAL_ATOMIC_MIN_U32); buffer initialized to -inf.
__device__ __forceinline__ void atomicMaxF(float* addr, float val) {
  if (val >= 0.0f) atomicMax((int*)addr, __float_as_int(val));
  else             atomicMin((unsigned int*)addr, __float_as_uint(val));
}

// ---------------------------------------------------------------------------
// Kernel 1: typed QKV projections.  One block = 16-node tile (uniform type),
// 8 waves = 8 output 16-col tiles.  A fragment shared across the 3 GEMMs.
// ---------------------------------------------------------------------------
__global__ void hgt_qkv_kernel(const float* __restrict__ x,
                               const float* __restrict__ Wk, const float* __restrict__ bk,
                               const float* __restrict__ Wq, const float* __restrict__ bq,
                               const float* __restrict__ Wv, const float* __restrict__ bv,
                               float* __restrict__ kout, float* __restrict__ qout,
                               float* __restrict__ vout) {
  const int tile = blockIdx.x;
  const int wave = threadIdx.x >> 5;
  const int lane = threadIdx.x & 31;
  const int l    = lane & 15;
  const int half = lane >> 4;
  const int out0 = wave * 16;
  const int t    = (tile * 16) / NT_SZ;
  const int row  = tile * 16 + l;                 // A-row handled by this lane

  const float* __restrict__ Wkt = Wk + t * DIM * DIM;
  const float* __restrict__ Wqt = Wq + t * DIM * DIM;
  const float* __restrict__ Wvt = Wv + t * DIM * DIM;

  v8f ck = {}, cq = {}, cv = {};
#pragma unroll 4
  for (int ks = 0; ks < DIM; ks += 4) {
    const int ka = ks + half * 2;
    v2f a;  a[0] = x[row * DIM + ka];            a[1] = x[row * DIM + ka + 1];
    v2f bK; bK[0] = Wkt[ka * DIM + out0 + l];    bK[1] = Wkt[(ka + 1) * DIM + out0 + l];
    v2f bQ; bQ[0] = Wqt[ka * DIM + out0 + l];    bQ[1] = Wqt[(ka + 1) * DIM + out0 + l];
    v2f bV; bV[0] = Wvt[ka * DIM + out0 + l];    bV[1] = Wvt[(ka + 1) * DIM + out0 + l];
    ck = wmma_f32(a, bK, ck);
    cq = wmma_f32(a, bQ, cq);
    cv = wmma_f32(a, bV, cv);
  }
  const float bsk = bk[t * DIM + out0 + l];
  const float bsq = bq[t * DIM + out0 + l];
  const float bsv = bv[t * DIM + out0 + l];
#pragma unroll
  for (int j = 0; j < 8; ++j) {
    const int rr = tile * 16 + j + half * 8;
    kout[rr * DIM + out0 + l] = ck[j] + bsk;
    qout[rr * DIM + out0 + l] = cq[j] + bsq;
    vout[rr * DIM + out0 + l] = cv[j] + bsv;
  }
}

// ---------------------------------------------------------------------------
// Kernel 2: edge logits.  One block = (relation r, 16-edge tile); wave = head.
// key_t = k[src] @ rel_att[r,h]  via one 16x16x16 WMMA group, then cross-lane
// dot with q[dst] + additive-attention term; atomic segment max.
// ---------------------------------------------------------------------------
__global__ void hgt_edge_logits_kernel(const float* __restrict__ kbuf,
                                       const float* __restrict__ qbuf,
                                       const int* __restrict__ src,
                                       const int* __restrict__ dst,
                                       const float* __restrict__ rel_att,
                                       const float* __restrict__ rel_pri,
                                       const float* __restrict__ nta,
                                       const float* __restrict__ nta1,
                                       const float* __restrict__ weight,
                                       const float* __restrict__ attn_w,
                                       float* __restrict__ sbuf,
                                       float* __restrict__ mbuf) {
  const int bx   = blockIdx.x;
  const int r    = bx / ETILES;
  const int e0   = (bx % ETILES) * 16;
  const int h    = threadIdx.x >> 5;
  const int lane = threadIdx.x & 31;
  const int l    = lane & 15;
  const int half = lane >> 4;

  const int   sA = src[r * EPR + e0 + l];         // A-row edge for this lane
  const float* __restrict__ B = rel_att + (r * H_HEADS + h) * DKK * DKK;

  v8f c = {};
#pragma unroll
  for (int ks = 0; ks < DKK; ks += 4) {
    const int ka = ks + half * 2;
    v2f a; a[0] = kbuf[sA * DIM + h * DKK + ka]; a[1] = kbuf[sA * DIM + h * DKK + ka + 1];
    v2f b; b[0] = B[ka * DKK + l];               b[1] = B[(ka + 1) * DKK + l];
    c = wmma_f32(a, b, c);
  }

  const float wq_l = attn_w[l];
  const float wk_l = attn_w[DKK + l];
  const float pri  = rel_pri[r * H_HEADS + h] * 0.25f;   // 1/sqrt(DK)
  const float beta = sigmoidf_(weight[0]);
  const float ns   = nta[src_nt(r)];
  const float nd   = nta1[dst_nt(r)];

#pragma unroll
  for (int j = 0; j < 8; ++j) {
    const int ej = e0 + j + half * 8;             // edge held in C[j] of this half
    const int sj = src[r * EPR + ej];
    const int dj = dst[r * EPR + ej];
    const float qv = qbuf[dj * DIM + h * DKK + l];
    const float kv = kbuf[sj * DIM + h * DKK + l];
    float sa = qv * c[j];                          // q . key_t
    float sq = qv * wq_l;                          // q . wq
    float sk = kv * wk_l;                          // k . wk
#pragma unroll
    for (int m = 1; m < 16; m <<= 1) {             // reduce within each half-wave
      sa += __shfl_xor(sa, m, 32);
      sq += __shfl_xor(sq, m, 32);
      sk += __shfl_xor(sk, m, 32);
    }
    if (l == 0) {
      const float pre = nd * sq + ns * sk;
      const float na  = pre > 0.0f ? pre : 0.01f * pre;   // leaky_relu
      const float s   = sa * pri + beta * na;
      sbuf[(r * EPR + ej) * H_HEADS + h] = s;
      atomicMaxF(&mbuf[(r * N_NODES + dj) * H_HEADS + h], s);
    }
  }
}

// ---------------------------------------------------------------------------
// Kernel 3: e = exp(s - m[seg]); den[seg] += e   (one thread per edge*head)
// ---------------------------------------------------------------------------
__global__ void hgt_exp_den_kernel(const float* __restrict__ mbuf,
                                   const int* __restrict__ dst,
                                   float* __restrict__ sbuf,
                                   float* __restrict__ den) {
  const int idx = blockIdx.x * blockDim.x + threadIdx.x;
  if (idx >= R_REL * EPR * H_HEADS) return;
  const int h  = idx & (H_HEADS - 1);
  const int el = idx >> 3;
  const int r  = el / EPR;
  const int e  = el % EPR;
  const int dj = dst[r * EPR + e];
  const float m  = mbuf[(r * N_NODES + dj) * H_HEADS + h];
  const float ev = expf(sbuf[idx] - m);
  sbuf[idx] = ev;
  atomicAdd(&den[(r * N_NODES + dj) * H_HEADS + h], ev);
}

// ---------------------------------------------------------------------------
// Kernel 4: val = v[src] @ rel_msg[r,h] (WMMA), scaled by att = e/den,
// atomically accumulated into acc[dst] (sum over relations is linear).
// ---------------------------------------------------------------------------
__global__ void hgt_msg_acc_kernel(const float* __restrict__ vbuf,
                                   const int* __restrict__ src,
                                   const int* __restrict__ dst,
                                   const float* __restrict__ rel_msg,
                                   const float* __restrict__ ebuf,
                                   const float* __restrict__ den,
                                   float* __restrict__ acc) {
  const int bx   = blockIdx.x;
  const int r    = bx / ETILES;
  const int e0   = (bx % ETILES) * 16;
  const int h    = threadIdx.x >> 5;
  const int lane = threadIdx.x & 31;
  const int l    = lane & 15;
  const int half = lane >> 4;

  const int   sA = src[r * EPR + e0 + l];
  const float* __restrict__ B = rel_msg + (r * H_HEADS + h) * DKK * DKK;

  v8f c = {};
#pragma unroll
  for (int ks = 0; ks < DKK; ks += 4) {
    const int ka = ks + half * 2;
    v2f a; a[0] = vbuf[sA * DIM + h * DKK + ka]; a[1] = vbuf[sA * DIM + h * DKK + ka + 1];
    v2f b; b[0] = B[ka * DKK + l];               b[1] = B[(ka + 1) * DKK + l];
    c = wmma_f32(a, b, c);
  }

#pragma unroll
  for (int j = 0; j < 8; ++j) {
    const int ej = e0 + j + half * 8;
    const int dj = dst[r * EPR + ej];
    const float ev  = ebuf[(r * EPR + ej) * H_HEADS + h];
    const float dn  = den[(r * N_NODES + dj) * H_HEADS + h];
    const float att = ev / dn;
    atomicAdd(&acc[dj * DIM + h * DKK + l], att * c[j]);
  }
}

// ---------------------------------------------------------------------------
// Kernel 5: nrel per node from den>0 (presence of each relation's segment)
// ---------------------------------------------------------------------------
__global__ void hgt_nrel_kernel(const float* __restrict__ den,
                                float* __restrict__ invn) {
  const int n = blockIdx.x * blockDim.x + threadIdx.x;
  if (n >= N_NODES) return;
  int cnt = 0;
#pragma unroll 2
  for (int r = 0; r < R_REL; ++r)
    cnt += (den[(r * N_NODES + n) * H_HEADS] > 0.0f) ? 1 : 0;
  invn[n] = 1.0f / fmaxf((float)cnt, 1.0f);
}

// ---------------------------------------------------------------------------
// Kernel 6: final typed GEMM  trans = (acc/nrel) @ Wa[t] + ba, skip blend.
// ---------------------------------------------------------------------------
__global__ void hgt_out_kernel(const float* __restrict__ acc,
                               const float* __restrict__ invn,
                               const float* __restrict__ x,
                               const float* __restrict__ Wa,
                               const float* __restrict__ ba,
                               const float* __restrict__ skip,
                               float* __restrict__ out) {
  const int tile = blockIdx.x;
  const int wave = threadIdx.x >> 5;
  const int lane = threadIdx.x & 31;
  const int l    = lane & 15;
  const int half = lane >> 4;
  const int out0 = wave * 16;
  const int t    = (tile * 16) / NT_SZ;
  const int row  = tile * 16 + l;
  const float sc = invn[row];                     // 1/nrel folded into A load

  const float* __restrict__ W = Wa + t * DIM * DIM;
  v8f c = {};
#pragma unroll 4
  for (int ks = 0; ks < DIM; ks += 4) {
    const int ka = ks + half * 2;
    v2f a; a[0] = acc[row * DIM + ka] * sc;  a[1] = acc[row * DIM + ka + 1] * sc;
    v2f b; b[0] = W[ka * DIM + out0 + l];    b[1] = W[(ka + 1) * DIM + out0 + l];
    c = wmma_f32(a, b, c);
  }
  const float bias  = ba[t * DIM + out0 + l];
  const float alpha = sigmoidf_(skip[t]);
#pragma unroll
  for (int j = 0; j < 8; ++j) {
    const int rr  = tile * 16 + j + half * 8;
    const int col = out0 + l;
    const float trans = c[j] + bias;
    out[rr * DIM + col] = trans * alpha + x[rr * DIM + col] * (1.0f - alpha);
  }
}

// ---------------------------------------------------------------------------
extern "C" void kernel_launch(void* const* d_in, const int* in_sizes, int n_in,
                              void* d_out, int out_size, void* d_ws, size_t ws_size,
                              hipStream_t stream) {
  (void)in_sizes; (void)n_in; (void)out_size; (void)ws_size;
  const float* x       = (const float*)d_in[0];
  const int*   src     = (const int*)  d_in[1];
  const int*   dst     = (const int*)  d_in[2];
  const float* Wk      = (const float*)d_in[3];
  const float* bk      = (const float*)d_in[4];
  const float* Wq      = (const float*)d_in[5];
  const float* bq      = (const float*)d_in[6];
  const float* Wv      = (const float*)d_in[7];
  const float* bv      = (const float*)d_in[8];
  const float* Wa      = (const float*)d_in[9];
  const float* ba      = (const float*)d_in[10];
  const float* rel_att = (const float*)d_in[11];
  const float* rel_msg = (const float*)d_in[12];
  const float* rel_pri = (const float*)d_in[13];
  const float* nta     = (const float*)d_in[14];
  const float* nta1    = (const float*)d_in[15];
  const float* skip    = (const float*)d_in[16];
  const float* weight  = (const float*)d_in[17];
  const float* attn_w  = (const float*)d_in[18];

  // workspace layout (floats)
  float* ws   = (float*)d_ws;
  float* kbuf = ws;                                  // N*DIM   = 3,072,000
  float* qbuf = kbuf + (size_t)N_NODES * DIM;
  float* vbuf = qbuf + (size_t)N_NODES * DIM;
  float* sbuf = vbuf + (size_t)N_NODES * DIM;        // R*EPR*H = 2,720,000
  float* mbuf = sbuf + (size_t)R_REL * EPR * H_HEADS;// R*N*H   = 6,528,000
  float* den  = mbuf + (size_t)R_REL * N_NODES * H_HEADS;
  float* acc  = den  + (size_t)R_REL * N_NODES * H_HEADS; // N*DIM
  float* invn = acc  + (size_t)N_NODES * DIM;        // N

  // init: m = -inf, den = 0, acc = 0   (graph-capturable async memsets)
  hipMemsetD32Async((hipDeviceptr_t)mbuf, (int)0xFF800000,
                    (size_t)R_REL * N_NODES * H_HEADS, stream);
  hipMemsetAsync(den, 0, (size_t)R_REL * N_NODES * H_HEADS * sizeof(float), stream);
  hipMemsetAsync(acc, 0, (size_t)N_NODES * DIM * sizeof(float), stream);

  hgt_qkv_kernel<<<NTILES, 256, 0, stream>>>(x, Wk, bk, Wq, bq, Wv, bv,
                                             kbuf, qbuf, vbuf);
  hgt_edge_logits_kernel<<<R_REL * ETILES, 256, 0, stream>>>(
      kbuf, qbuf, src, dst, rel_att, rel_pri, nta, nta1, weight, attn_w,
      sbuf, mbuf);
  hgt_exp_den_kernel<<<(R_REL * EPR * H_HEADS + 255) / 256, 256, 0, stream>>>(
      mbuf, dst, sbuf, den);
  hgt_msg_acc_kernel<<<R_REL * ETILES, 256, 0, stream>>>(
      vbuf, src, dst, rel_msg, sbuf, den, acc);
  hgt_nrel_kernel<<<(N_NODES + 255) / 256, 256, 0, stream>>>(den, invn);
  hgt_out_kernel<<<NTILES, 256, 0, stream>>>(acc, invn, x, Wa, ba, skip,
                                             (float*)d_out);
}